// SharedBottomFramework_82025285419694
// MI455X (gfx1250) — compile-verified
//
#include <hip/hip_runtime.h>

// Problem dims (match reference)
#define F_   26
#define V_   100000
#define E_   16
#define B_   16384
#define ND_  13
#define D_   4
#define H_   128
#define T_   8
#define IN_  429               // F*E + ND
#define INP_ 432               // padded to multiple of K=4
#define KSTEPS (INP_ / 4)      // 108
#define NTILES 8               // 128 cols / 16
#define WAVES  4
#define NT_PER_WAVE (NTILES / WAVES)   // 2
#define BLOCK  (WAVES * 32)            // 128 threads

typedef __attribute__((ext_vector_type(2))) float v2f;
typedef __attribute__((ext_vector_type(4))) float v4f;
typedef __attribute__((ext_vector_type(8))) float v8f;
typedef __attribute__((ext_vector_type(4))) int   v4i;

// int4 in explicit address spaces for the async-to-LDS builtin
typedef __attribute__((address_space(1))) v4i as1_v4i;   // global
typedef __attribute__((address_space(3))) v4i as3_v4i;   // LDS

#define WS_PACK_ELEMS (KSTEPS * NTILES * 32)                 // 27648 b64 slots
#define WS_PACK_BYTES ((size_t)WS_PACK_ELEMS * sizeof(v2f))  // 221184 B

static __device__ __forceinline__ v8f wmma_f32(v2f a, v2f b, v8f c) {
    return __builtin_amdgcn_wmma_f32_16x16x4_f32(
        /*neg_a=*/false, a, /*neg_b=*/false, b,
        /*c_mod=*/(short)0, c, /*reuse_a=*/false, /*reuse_b=*/false);
}

// ---- One-shot repack of W_bottom [429,128] into per-lane WMMA B-fragment
// layout Wp[ks][nt][lane] = {W[ka][n], W[ka+1][n]}, zero-padded K rows baked in.
__global__ __launch_bounds__(256) void pack_w_kernel(const float* __restrict__ W,
                                                     v2f* __restrict__ Wp) {
    const int idx = blockIdx.x * 256 + threadIdx.x;
    if (idx >= WS_PACK_ELEMS) return;
    const int lane = idx & 31;
    const int t    = idx >> 5;
    const int nt   = t & 7;
    const int ks   = t >> 3;
    const int n    = nt * 16 + (lane & 15);
    const int ka   = ks * 4 + ((lane & 16) ? 2 : 0);
    v2f b;
    b.x = (ka     < IN_) ? W[(size_t)ka       * H_ + n] : 0.0f;
    b.y = (ka + 1 < IN_) ? W[(size_t)(ka + 1) * H_ + n] : 0.0f;
    Wp[idx] = b;
}

// ---- Fused main kernel: 4 waves / block, one 16-row batch tile per block.
// gather(async)->LDS, WMMA GEMM (2 N-tiles per wave), per-domain tower, sigmoid.
template <bool PACKED>
__global__ __launch_bounds__(BLOCK) void shared_bottom_fused(
    const int*   __restrict__ sparse_ids,   // [B, F]
    const float* __restrict__ dense,        // [B, ND]
    const int*   __restrict__ domain_id,    // [B]
    const float* __restrict__ emb_tables,   // [F, V, E]
    const float* __restrict__ W_bottom,     // [IN, H]
    const float* __restrict__ b_bottom,     // [H]
    const float* __restrict__ tower_W1,     // [D, H, T]
    const float* __restrict__ tower_b1,     // [D, T]
    const float* __restrict__ tower_W2,     // [D, T]
    const float* __restrict__ tower_b2,     // [D]
    const v2f*   __restrict__ Wp,           // packed B frags (d_ws) or null
    float*       __restrict__ out)          // [B]
{
    __shared__ float lds_x[16 * INP_];      // X tile [16][432]  (27.6 KB)
    __shared__ float lds_feat[H_ * 16];     // feat TRANSPOSED [n][m] (8 KB)

    const int tid  = threadIdx.x;
    const int lane = tid & 31;
    const int wv   = tid >> 5;
    const int m0   = blockIdx.x * 16;

    // ---- Stage 1: gather 16x26 embeddings as 16B chunks into LDS ----
    constexpr int CH = 16 * F_ * 4;         // 1664 chunks (divisible by BLOCK)
    for (int c = tid; c < CH; c += BLOCK) {
        const int r   = c / (F_ * 4);
        const int rem = c - r * (F_ * 4);
        const int f   = rem >> 2;
        const int q   = rem & 3;
        const int id  = sparse_ids[(m0 + r) * F_ + f];
        const float* src = emb_tables + ((size_t)f * V_ + (size_t)id) * E_ + q * 4;
        float*       dst = &lds_x[r * INP_ + f * E_ + q * 4];
#if __has_builtin(__builtin_amdgcn_global_load_async_to_lds_b128)
        __builtin_amdgcn_global_load_async_to_lds_b128(
            (as1_v4i*)src, (as3_v4i*)dst, /*offset=*/0, /*cpol=*/0);
#else
        *(v4f*)dst = *(const v4f*)src;
#endif
    }
    for (int c = tid; c < 16 * ND_; c += BLOCK) {
        const int r = c / ND_, cc = c - r * ND_;
        lds_x[r * INP_ + F_ * E_ + cc] = dense[(m0 + r) * ND_ + cc];
    }
    for (int c = tid; c < 16 * 3; c += BLOCK) {
        const int r = c / 3, cc = c - r * 3;
        lds_x[r * INP_ + IN_ + cc] = 0.0f;   // zero K padding cols 429..431
    }
#if __has_builtin(__builtin_amdgcn_global_load_async_to_lds_b128)
#if __has_builtin(__builtin_amdgcn_s_wait_asynccnt)
    __builtin_amdgcn_s_wait_asynccnt(0);
#else
    asm volatile("s_wait_asynccnt 0" ::: "memory");
#endif
#endif
    __syncthreads();

    // ---- Stage 2: feat = relu(X @ W_bottom + b) via V_WMMA_F32_16X16X4_F32 ----
    // A frag: lanes 0-15 -> M=lane, K=k0..k0+1 ; lanes 16-31 -> M=lane-16, K=k0+2..k0+3
    const int  l15 = lane & 15;
    const bool hi  = (lane & 16) != 0;
    const int  nt0 = wv * NT_PER_WAVE;

    v8f acc0 = {}, acc1 = {};
    const float* xk = &lds_x[l15 * INP_] + (hi ? 2 : 0);

    if constexpr (PACKED) {
        // One coalesced b64 per B-fragment, no guards anywhere.
        const v2f* wp = Wp + nt0 * 32 + lane;
        #pragma unroll 4
        for (int ks = 0; ks < KSTEPS; ++ks) {
            v2f a;
            a.x = xk[ks * 4];
            a.y = xk[ks * 4 + 1];
            const v2f b0 = wp[0];
            const v2f b1 = wp[32];
            acc0 = wmma_f32(a, b0, acc0);
            acc1 = wmma_f32(a, b1, acc1);
            wp += NTILES * 32;
        }
    } else {
        // Direct-W path: branch-free main loop (k<=427 is always in range),
        // guarded epilogue handles the 3 padded K rows.
        const int n0 = nt0 * 16 + l15;
        #pragma unroll 4
        for (int ks = 0; ks < KSTEPS - 1; ++ks) {
            const int k0 = ks * 4 + (hi ? 2 : 0);
            v2f a;
            a.x = xk[ks * 4];
            a.y = xk[ks * 4 + 1];
            const float* w0 = W_bottom + (size_t)k0 * H_ + n0;
            v2f b0, b1;
            b0.x = w0[0];        b0.y = w0[H_];
            b1.x = w0[16];       b1.y = w0[H_ + 16];
            acc0 = wmma_f32(a, b0, acc0);
            acc1 = wmma_f32(a, b1, acc1);
        }
        {
            const int ks = KSTEPS - 1;
            const int k0 = ks * 4 + (hi ? 2 : 0);
            v2f a;
            a.x = xk[ks * 4];
            a.y = xk[ks * 4 + 1];
            const bool g0 = (k0     < IN_);
            const bool g1 = (k0 + 1 < IN_);
            v2f b0, b1;
            b0.x = g0 ? W_bottom[(size_t)k0       * H_ + n0]      : 0.0f;
            b0.y = g1 ? W_bottom[(size_t)(k0 + 1) * H_ + n0]      : 0.0f;
            b1.x = g0 ? W_bottom[(size_t)k0       * H_ + n0 + 16] : 0.0f;
            b1.y = g1 ? W_bottom[(size_t)(k0 + 1) * H_ + n0 + 16] : 0.0f;
            acc0 = wmma_f32(a, b0, acc0);
            acc1 = wmma_f32(a, b1, acc1);
        }
    }

    // C layout: VGPR j -> row (j + 8*hi), lane&15 -> col. Bias + ReLU, store
    // transposed [n][m] so tower column reads hit 16 distinct LDS banks.
    #pragma unroll
    for (int i = 0; i < NT_PER_WAVE; ++i) {
        const v8f a = i ? acc1 : acc0;
        const int n = (nt0 + i) * 16 + l15;
        const float bb = b_bottom[n];
        #pragma unroll
        for (int j = 0; j < 8; ++j) {
            const int m = j + (hi ? 8 : 0);
            const float v = a[j] + bb;
            lds_feat[n * 16 + m] = v > 0.0f ? v : 0.0f;
        }
    }
    __syncthreads();

    // ---- Stage 3: per-domain tower (128 -> 8 -> 1) + sigmoid, one lane per row ----
    if (tid < 16) {
        const int row = m0 + tid;
        const int d   = domain_id[row];
        const float* W1 = tower_W1 + (size_t)d * H_ * T_;

        float s[T_];
        #pragma unroll
        for (int t = 0; t < T_; ++t) s[t] = tower_b1[d * T_ + t];

        #pragma unroll 4
        for (int h = 0; h < H_; ++h) {
            const float fv = lds_feat[h * 16 + tid];
            #pragma unroll
            for (int t = 0; t < T_; ++t) s[t] += fv * W1[h * T_ + t];
        }

        float logit = tower_b2[d];
        #pragma unroll
        for (int t = 0; t < T_; ++t) {
            const float ht = s[t] > 0.0f ? s[t] : 0.0f;
            logit += ht * tower_W2[d * T_ + t];
        }
        out[row] = 1.0f / (1.0f + __expf(-logit));
    }
}

extern "C" void kernel_launch(void* const* d_in, const int* in_sizes, int n_in,
                              void* d_out, int out_size, void* d_ws, size_t ws_size,
                              hipStream_t stream) {
    (void)in_sizes; (void)n_in; (void)out_size;
    const int*   sparse_ids = (const int*)  d_in[0];
    const float* dense      = (const float*)d_in[1];
    const int*   domain_id  = (const int*)  d_in[2];
    const float* emb_tables = (const float*)d_in[3];
    const float* W_bottom   = (const float*)d_in[4];
    const float* b_bottom   = (const float*)d_in[5];
    const float* tower_W1   = (const float*)d_in[6];
    const float* tower_b1   = (const float*)d_in[7];
    const float* tower_W2   = (const float*)d_in[8];
    const float* tower_b2   = (const float*)d_in[9];
    float*       out        = (float*)d_out;

    const bool packed = (d_ws != nullptr) && (ws_size >= WS_PACK_BYTES);
    if (packed) {
        pack_w_kernel<<<(WS_PACK_ELEMS + 255) / 256, 256, 0, stream>>>(
            W_bottom, (v2f*)d_ws);
        shared_bottom_fused<true><<<dim3(B_ / 16), dim3(BLOCK), 0, stream>>>(
            sparse_ids, dense, domain_id, emb_tables, W_bottom, b_bottom,
            tower_W1, tower_b1, tower_W2, tower_b2, (const v2f*)d_ws, out);
    } else {
        shared_bottom_fused<false><<<dim3(B_ / 16), dim3(BLOCK), 0, stream>>>(
            sparse_ids, dense, domain_id, emb_tables, W_bottom, b_bottom,
            tower_W1, tower_b1, tower_W2, tower_b2, nullptr, out);
    }
}